// MoDLayer_36172214567086
// MI455X (gfx1250) — compile-verified
//
#include <hip/hip_runtime.h>
#include <math.h>

typedef float v2f __attribute__((ext_vector_type(2)));
typedef float v8f __attribute__((ext_vector_type(8)));

#define NTOK   16384        // B*T
#define DMODEL 1024
#define DFF    4096
#define CAP    8192         // capacity = NTOK/2

#define BM   128            // block tile M
#define BN   256            // block tile N
#define KB   32             // K chunk per stage
#define LDA  36             // A lds row stride (pad)
#define LDB  264            // B lds row stride (pad)

__device__ __forceinline__ float sigmoid_precise(float x) {
    return 1.0f / (1.0f + expf(-x));
}
__device__ __forceinline__ float gelu_exact(float x) {
    return 0.5f * x * (1.0f + erff(x * 0.70710678118654752f));
}
__device__ __forceinline__ unsigned float_key(float f) {
    unsigned u = __float_as_uint(f);
    return (u & 0x80000000u) ? ~u : (u | 0x80000000u);
}
// Low 32 bits of a flat shared-memory address == LDS byte offset (aperture
// base has zero low word; HW adds LDS_BASE itself).
__device__ __forceinline__ unsigned lds_addr_of(const void* p) {
    return (unsigned)(uintptr_t)p;
}
// CDNA5 async global->LDS copy, 16 bytes per lane. Tracked by ASYNCcnt.
__device__ __forceinline__ void async_copy16(unsigned lds_addr, const float* gsrc) {
    asm volatile("global_load_async_to_lds_b128 %0, %1, off"
                 :: "v"(lds_addr), "v"(gsrc)
                 : "memory");
}
__device__ __forceinline__ void wait_async0() {
    asm volatile("s_wait_asynccnt 0x0" ::: "memory");
}

// ---------------------------------------------------------------------------
__global__ void init_kernel(unsigned* uctl) {
    uctl[2] = 0u;  // cnt_sel
    uctl[3] = 0u;  // cnt_eq
}

// ---------------------------------------------------------------------------
// Router scores: one wave per token; also copies hidden -> out.
__global__ __launch_bounds__(256) void router_kernel(
    const float* __restrict__ hidden, const float* __restrict__ wr,
    float* __restrict__ out, float* __restrict__ scores,
    unsigned* __restrict__ keys) {
    const int tok  = (blockIdx.x * 256 + threadIdx.x) >> 5;
    const int lane = threadIdx.x & 31;
    const float* row  = hidden + (size_t)tok * DMODEL;
    float*       orow = out    + (size_t)tok * DMODEL;
    float acc = 0.0f;
#pragma unroll
    for (int i = 0; i < 8; ++i) {
        const int c = i * 128 + lane * 4;
        float4 hv = *(const float4*)(row + c);
        float4 wv = *(const float4*)(wr + c);
        *(float4*)(orow + c) = hv;
        acc += hv.x * wv.x + hv.y * wv.y + hv.z * wv.z + hv.w * wv.w;
    }
#pragma unroll
    for (int off = 16; off > 0; off >>= 1) acc += __shfl_xor(acc, off, 32);
    if (lane == 0) {
        scores[tok] = acc;
        keys[tok]   = float_key(acc);
    }
}

// ---------------------------------------------------------------------------
// 4-pass radix select of the k-th largest key. uctl[0]=threshold, uctl[1]=ties.
__global__ __launch_bounds__(256) void select_kernel(
    const unsigned* __restrict__ keys, unsigned* __restrict__ uctl) {
    __shared__ unsigned hist[256];
    __shared__ unsigned sh_prefix, sh_k;
    const int tid = threadIdx.x;
    if (tid == 0) { sh_prefix = 0u; sh_k = (unsigned)CAP; }
    __syncthreads();
    for (int pass = 0; pass < 4; ++pass) {
        const int shift = 24 - pass * 8;
        hist[tid] = 0u;
        __syncthreads();
        const unsigned prefix  = sh_prefix;
        const unsigned mask_hi = (pass == 0) ? 0u : (0xFFFFFFFFu << (shift + 8));
        for (int i = tid; i < NTOK; i += 256) {
            const unsigned key = keys[i];
            if ((key & mask_hi) == prefix)
                atomicAdd(&hist[(key >> shift) & 255u], 1u);
        }
        __syncthreads();
        if (tid == 0) {
            unsigned kk = sh_k;
            for (int b = 255; b >= 0; --b) {
                const unsigned c = hist[b];
                if (kk <= c) {
                    sh_prefix = prefix | ((unsigned)b << shift);
                    sh_k = kk;
                    break;
                }
                kk -= c;
            }
        }
        __syncthreads();
    }
    if (tid == 0) { uctl[0] = sh_prefix; uctl[1] = sh_k; }
}

// ---------------------------------------------------------------------------
__global__ __launch_bounds__(256) void compact_kernel(
    const unsigned* __restrict__ keys, const float* __restrict__ scores,
    unsigned* __restrict__ uctl, int* __restrict__ sel, float* __restrict__ rw) {
    const int i = blockIdx.x * 256 + threadIdx.x;
    if (i >= NTOK) return;
    const unsigned key = keys[i];
    const unsigned thr = uctl[0];
    const unsigned rem = uctl[1];
    if (key > thr) {
        const unsigned pos = atomicAdd(&uctl[2], 1u);
        sel[pos] = i;
        rw[pos]  = sigmoid_precise(scores[i]);
    } else if (key == thr) {
        const unsigned e = atomicAdd(&uctl[3], 1u);
        if (e < rem) {
            const unsigned pos = atomicAdd(&uctl[2], 1u);
            sel[pos] = i;
            rw[pos]  = sigmoid_precise(scores[i]);
        }
    }
}

// ---------------------------------------------------------------------------
// FFN layer 1: h = gelu(gather(hidden, sel) @ w1 + b1)   (CAP x DFF)
// 128x256 block tile; 8 waves (2x4) each computing 64x64 via fp32 WMMA.
// Double-buffered LDS fed by async global->LDS copies.
__global__ __launch_bounds__(256) void ffn1_kernel(
    const float* __restrict__ hidden, const float* __restrict__ w1,
    const float* __restrict__ b1, const int* __restrict__ sel,
    float* __restrict__ hbuf) {
    __shared__ float Alds[2][BM * LDA];
    __shared__ float Blds[2][KB * LDB];
    __shared__ int   selrow[BM];

    const int tid  = threadIdx.x;
    const int lane = tid & 31;
    const int w    = tid >> 5;
    const int wm   = w >> 2;      // 2 waves along M (64 rows each)
    const int wn   = w & 3;       // 4 waves along N (64 cols each)
    const int lm   = lane & 15;
    const int half = lane >> 4;
    const int bm = blockIdx.x, bn = blockIdx.y;

    if (tid < BM) selrow[tid] = sel[bm * BM + tid];
    __syncthreads();

    const v8f zero = {0.f, 0.f, 0.f, 0.f, 0.f, 0.f, 0.f, 0.f};
    v8f c[4][4];
#pragma unroll
    for (int mt = 0; mt < 4; ++mt)
#pragma unroll
        for (int nt = 0; nt < 4; ++nt) c[mt][nt] = zero;

    auto issue_stage = [&](int kk0, int buf) {
#pragma unroll
        for (int i = 0; i < 4; ++i) {           // A: 128 x 32 gathered
            const int idx = tid + i * 256;
            const int r = idx >> 3, c4 = (idx & 7) << 2;
            const float* src = hidden + (size_t)selrow[r] * DMODEL + kk0 + c4;
            async_copy16(lds_addr_of(&Alds[buf][r * LDA + c4]), src);
        }
#pragma unroll
        for (int i = 0; i < 8; ++i) {           // B: 32 x 256
            const int idx = tid + i * 256;
            const int r = idx >> 6, c4 = (idx & 63) << 2;
            const float* src = w1 + (size_t)(kk0 + r) * DFF + bn * BN + c4;
            async_copy16(lds_addr_of(&Blds[buf][r * LDB + c4]), src);
        }
    };

    const int NSTAGE = DMODEL / KB;
    issue_stage(0, 0);
    for (int s = 0; s < NSTAGE; ++s) {
        const int cur = s & 1;
        wait_async0();
        __syncthreads();
        if (s + 1 < NSTAGE) issue_stage((s + 1) * KB, 1 - cur);
        const float* Ab = Alds[cur];
        const float* Bb = Blds[cur];
#pragma unroll
        for (int ks = 0; ks < KB; ks += 4) {
            const int koff = ks + half * 2;
            v2f a[4], b[4];
#pragma unroll
            for (int mt = 0; mt < 4; ++mt)
                a[mt] = *(const v2f*)&Ab[(wm * 64 + mt * 16 + lm) * LDA + koff];
#pragma unroll
            for (int nt = 0; nt < 4; ++nt) {
                const int n = wn * 64 + nt * 16 + lm;
                b[nt].x = Bb[koff * LDB + n];
                b[nt].y = Bb[(koff + 1) * LDB + n];
            }
#pragma unroll
            for (int mt = 0; mt < 4; ++mt)
#pragma unroll
                for (int nt = 0; nt < 4; ++nt)
                    c[mt][nt] = __builtin_amdgcn_wmma_f32_16x16x4_f32(
                        false, a[mt], false, b[nt], (short)0, c[mt][nt], false, false);
        }
        __syncthreads();
    }
#pragma unroll
    for (int mt = 0; mt < 4; ++mt) {
#pragma unroll
        for (int nt = 0; nt < 4; ++nt) {
            const int ng = bn * BN + wn * 64 + nt * 16 + lm;
            const float bias = b1[ng];
#pragma unroll
            for (int r = 0; r < 8; ++r) {
                const int mg = bm * BM + wm * 64 + mt * 16 + half * 8 + r;
                const float x = c[mt][nt][r] + bias;
                hbuf[(size_t)mg * DFF + ng] = gelu_exact(x);
            }
        }
    }
}

// ---------------------------------------------------------------------------
// FFN layer 2: out[sel[m]] = (h @ w2 + b2) * rw[m]   (CAP x DMODEL scatter)
__global__ __launch_bounds__(256) void ffn2_kernel(
    const float* __restrict__ hbuf, const float* __restrict__ w2,
    const float* __restrict__ b2, const int* __restrict__ sel,
    const float* __restrict__ rw, float* __restrict__ out) {
    __shared__ float Alds[2][BM * LDA];
    __shared__ float Blds[2][KB * LDB];
    __shared__ int   selrow[BM];
    __shared__ float rwrow[BM];

    const int tid  = threadIdx.x;
    const int lane = tid & 31;
    const int w    = tid >> 5;
    const int wm   = w >> 2;
    const int wn   = w & 3;
    const int lm   = lane & 15;
    const int half = lane >> 4;
    const int bm = blockIdx.x, bn = blockIdx.y;

    if (tid < BM) {
        selrow[tid] = sel[bm * BM + tid];
        rwrow[tid]  = rw[bm * BM + tid];
    }
    __syncthreads();

    const v8f zero = {0.f, 0.f, 0.f, 0.f, 0.f, 0.f, 0.f, 0.f};
    v8f c[4][4];
#pragma unroll
    for (int mt = 0; mt < 4; ++mt)
#pragma unroll
        for (int nt = 0; nt < 4; ++nt) c[mt][nt] = zero;

    auto issue_stage = [&](int kk0, int buf) {
#pragma unroll
        for (int i = 0; i < 4; ++i) {           // A: 128 x 32 from hbuf
            const int idx = tid + i * 256;
            const int r = idx >> 3, c4 = (idx & 7) << 2;
            const float* src = hbuf + (size_t)(bm * BM + r) * DFF + kk0 + c4;
            async_copy16(lds_addr_of(&Alds[buf][r * LDA + c4]), src);
        }
#pragma unroll
        for (int i = 0; i < 8; ++i) {           // B: 32 x 256 from w2
            const int idx = tid + i * 256;
            const int r = idx >> 6, c4 = (idx & 63) << 2;
            const float* src = w2 + (size_t)(kk0 + r) * DMODEL + bn * BN + c4;
            async_copy16(lds_addr_of(&Blds[buf][r * LDB + c4]), src);
        }
    };

    const int NSTAGE = DFF / KB;
    issue_stage(0, 0);
    for (int s = 0; s < NSTAGE; ++s) {
        const int cur = s & 1;
        wait_async0();
        __syncthreads();
        if (s + 1 < NSTAGE) issue_stage((s + 1) * KB, 1 - cur);
        const float* Ab = Alds[cur];
        const float* Bb = Blds[cur];
#pragma unroll
        for (int ks = 0; ks < KB; ks += 4) {
            const int koff = ks + half * 2;
            v2f a[4], b[4];
#pragma unroll
            for (int mt = 0; mt < 4; ++mt)
                a[mt] = *(const v2f*)&Ab[(wm * 64 + mt * 16 + lm) * LDA + koff];
#pragma unroll
            for (int nt = 0; nt < 4; ++nt) {
                const int n = wn * 64 + nt * 16 + lm;
                b[nt].x = Bb[koff * LDB + n];
                b[nt].y = Bb[(koff + 1) * LDB + n];
            }
#pragma unroll
            for (int mt = 0; mt < 4; ++mt)
#pragma unroll
                for (int nt = 0; nt < 4; ++nt)
                    c[mt][nt] = __builtin_amdgcn_wmma_f32_16x16x4_f32(
                        false, a[mt], false, b[nt], (short)0, c[mt][nt], false, false);
        }
        __syncthreads();
    }
#pragma unroll
    for (int mt = 0; mt < 4; ++mt) {
#pragma unroll
        for (int nt = 0; nt < 4; ++nt) {
            const int ng = bn * BN + wn * 64 + nt * 16 + lm;
            const float bias = b2[ng];
#pragma unroll
            for (int r = 0; r < 8; ++r) {
                const int ml = wm * 64 + mt * 16 + half * 8 + r;
                const float v = (c[mt][nt][r] + bias) * rwrow[ml];
                out[(size_t)selrow[ml] * DMODEL + ng] = v;
            }
        }
    }
}

// ---------------------------------------------------------------------------
// Load-balance loss, deterministic fixed-order reduction.
__global__ __launch_bounds__(256) void loss_kernel(
    const float* __restrict__ scores, float* __restrict__ out_loss) {
    __shared__ float ps[256], ps2[256];
    const int tid = threadIdx.x;
    float s = 0.f, s2 = 0.f;
    for (int i = tid; i < NTOK; i += 256) {
        const float p = sigmoid_precise(scores[i]);
        s += p;
        s2 += p * p;
    }
    ps[tid] = s;
    ps2[tid] = s2;
    __syncthreads();
    if (tid == 0) {
        float S = 0.f, S2 = 0.f;
        for (int i = 0; i < 256; ++i) { S += ps[i]; S2 += ps2[i]; }
        const float n = (float)NTOK;
        const float mean = S / n;
        const float var  = (S2 - S * S / n) / (n - 1.0f);
        *out_loss = var + (mean - 0.5f) * (mean - 0.5f);
    }
}

// ---------------------------------------------------------------------------
extern "C" void kernel_launch(void* const* d_in, const int* in_sizes, int n_in,
                              void* d_out, int out_size, void* d_ws, size_t ws_size,
                              hipStream_t stream) {
    (void)in_sizes; (void)n_in; (void)out_size; (void)ws_size;
    const float* hidden = (const float*)d_in[0];
    const float* wr     = (const float*)d_in[1];
    const float* w1     = (const float*)d_in[2];
    const float* b1     = (const float*)d_in[3];
    const float* w2     = (const float*)d_in[4];
    const float* b2     = (const float*)d_in[5];
    float* out = (float*)d_out;

    // Workspace layout
    float*    scores = (float*)d_ws;                      // NTOK
    unsigned* keys   = (unsigned*)(scores + NTOK);        // NTOK
    int*      sel    = (int*)(keys + NTOK);               // CAP
    float*    rw     = (float*)(sel + CAP);               // CAP
    unsigned* uctl   = (unsigned*)(rw + CAP);             // [thr, rem, cnt_sel, cnt_eq]
    float*    hbuf   = (float*)((char*)d_ws + (1 << 18)); // CAP x DFF f32 (128 MB)

    init_kernel<<<1, 1, 0, stream>>>(uctl);
    router_kernel<<<NTOK * 32 / 256, 256, 0, stream>>>(hidden, wr, out, scores, keys);
    select_kernel<<<1, 256, 0, stream>>>(keys, uctl);
    compact_kernel<<<NTOK / 256, 256, 0, stream>>>(keys, scores, uctl, sel, rw);
    ffn1_kernel<<<dim3(CAP / BM, DFF / BN), 256, 0, stream>>>(hidden, w1, b1, sel, hbuf);
    ffn2_kernel<<<dim3(CAP / BM, DMODEL / BN), 256, 0, stream>>>(hbuf, w2, b2, sel, rw, out);
    loss_kernel<<<1, 256, 0, stream>>>(scores, out + (size_t)NTOK * DMODEL);
}